// dynamic_unfolding_32117765439959
// MI455X (gfx1250) — compile-verified
//
#include <hip/hip_runtime.h>

// ---------------------------------------------------------------------------
// dynamic_unfolding for MI455X (gfx1250)
//
// Shapes (fixed by setup_inputs): x [8,128,256,256] f32, local_count
// [8,1,249,249] f32, k = 8. Output [8,1,256,256] f32.
//
// Roofline: x = 268 MB read dominates (~11.5 us @ 23.3 TB/s); all other
// tensors ~2 MB (L2-resident, 192 MB L2). So: one bandwidth-optimal streaming
// pass computing the channel mean (as a WMMA f32 GEMM with a constant
// 1/128 A-matrix), fused with exp(); then two tiny box-filter passes:
//   S[b,i,j]  = sum_{8x8} E[b,i+di,j+dj]         (window softmax denominator)
//   W[b,i,j]  = local_count[b,i,j] / S[b,i,j]
//   R[b,y,x]  = E[b,y,x] * sum_{windows covering (y,x)} W[b,i,j]
// which is exactly softmax * count, fold-scatter-added.
// ---------------------------------------------------------------------------

typedef __attribute__((ext_vector_type(2))) float v2f;
typedef __attribute__((ext_vector_type(8))) float v8f;

#define NB   8
#define NC   128
#define HH   256
#define WW   256
#define HW   (HH * WW)        // 65536
#define KW   8
#define HP   (HH - KW + 1)    // 249
#define WP   (WW - KW + 1)    // 249

#if __has_builtin(__builtin_amdgcn_wmma_f32_16x16x4_f32)
#define USE_WMMA_F32X4 1
#else
#define USE_WMMA_F32X4 0
#endif

// ---------------------------------------------------------------------------
// Kernel 1: E[b,p] = exp( mean_c x[b,c,p] )   -- the 268 MB streaming pass.
// WMMA formulation: D = A * B + C with A = const 1/128 (16x4), B = 4 channels
// x 16 pixels of x. 32 chained WMMAs reduce all 128 channels. With a constant
// A the layout/permutation details of A and the K-slot assignment of B cancel,
// and B/D share the same lane->N striping, so lane `col` always ends up with
// the mean of the pixel it loaded. Two chains per wave (32 pixels) for ILP.
// EXEC stays all-ones: exact grid, no divergence.
// ---------------------------------------------------------------------------
__global__ __launch_bounds__(256) void mean_exp_kernel(const float* __restrict__ x,
                                                       float* __restrict__ E) {
  const int    tid = threadIdx.x;
  const size_t blk = (size_t)blockIdx.x * 256;   // 256 pixels per block
  const int    b   = (int)(blk >> 16);           // HW = 65536
  const float* xb  = x + (size_t)b * NC * HW;
#if USE_WMMA_F32X4
  const int lane = tid & 31;
  const int wv   = tid >> 5;        // wave32: 8 waves/block
  const int col  = lane & 15;
  const int hi   = lane >> 4;       // half-wave select
  const int pb   = (int)(blk & (HW - 1)) + wv * 32;

  v8f acc0 = {};
  v8f acc1 = {};
  v2f a;
  a.x = 1.0f / 128.0f;
  a.y = 1.0f / 128.0f;              // constant A matrix => layout-agnostic

  // lane's two B components = channels (c0+2*hi) and (c0+2*hi+1), pixel pb+col
  const float* p = xb + (size_t)(2 * hi) * HW + (size_t)(pb + col);
#pragma unroll 4
  for (int c0 = 0; c0 < NC; c0 += 4) {
    v2f b0, b1;
    b0.x = p[0];            // chain0: pixels pb+0..15
    b0.y = p[HW];
    b1.x = p[16];           // chain1: pixels pb+16..31
    b1.y = p[HW + 16];
    p += (size_t)4 * HW;
    acc0 = __builtin_amdgcn_wmma_f32_16x16x4_f32(false, a, false, b0,
                                                 (short)0, acc0, false, false);
    acc1 = __builtin_amdgcn_wmma_f32_16x16x4_f32(false, a, false, b1,
                                                 (short)0, acc1, false, false);
  }
  // All 16 rows of D are equal; row 0/8 lives in VGPR0 with N = lane&15.
  const float m = hi ? acc1[0] : acc0[0];
  E[(size_t)b * HW + (size_t)(pb + hi * 16 + col)] = expf(m);
#else
  // Fallback (also compiled on the host pass where the builtin is absent):
  // plain coalesced per-thread channel reduction.
  const int    pp = (int)(blk & (HW - 1)) + tid;
  const float* p  = xb + pp;
  float s = 0.0f;
#pragma unroll 8
  for (int c = 0; c < NC; ++c) s += p[(size_t)c * HW];
  E[(size_t)b * HW + pp] = expf(s * (1.0f / 128.0f));
#endif
}

// ---------------------------------------------------------------------------
// Kernel 2: per-window softmax denominator + count scaling (L2-resident).
//   Wgt[b,i,j] = local_count[b,i,j] / sum_{di,dj<8} E[b,i+di,j+dj]
// ---------------------------------------------------------------------------
__global__ __launch_bounds__(256) void window_weight_kernel(const float* __restrict__ E,
                                                            const float* __restrict__ lc,
                                                            float* __restrict__ Wgt) {
  const int idx = blockIdx.x * 256 + threadIdx.x;
  if (idx >= NB * HP * WP) return;
  const int b = idx / (HP * WP);
  const int r = idx % (HP * WP);
  const int i = r / WP;
  const int j = r % WP;
  const float* Eb = E + (size_t)b * HW + (size_t)i * WW + j;
  float s = 0.0f;
#pragma unroll
  for (int di = 0; di < KW; ++di) {
#pragma unroll
    for (int dj = 0; dj < KW; ++dj) s += Eb[di * WW + dj];
  }
  Wgt[idx] = lc[idx] / s;
}

// ---------------------------------------------------------------------------
// Kernel 3: fold (overlap-add as a gather over covering windows).
//   R[b,y,x] = E[b,y,x] * sum_{i in [y-7,y] clamped, j in [x-7,x] clamped} Wgt
// ---------------------------------------------------------------------------
__global__ __launch_bounds__(256) void fold_kernel(const float* __restrict__ E,
                                                   const float* __restrict__ Wgt,
                                                   float* __restrict__ out) {
  const int idx = blockIdx.x * 256 + threadIdx.x;
  if (idx >= NB * HW) return;
  const int b  = idx >> 16;
  const int r  = idx & (HW - 1);
  const int y  = r >> 8;
  const int xx = r & 255;
  const int i0 = (y - 7 > 0) ? (y - 7) : 0;
  const int i1 = (y < HP - 1) ? y : (HP - 1);
  const int j0 = (xx - 7 > 0) ? (xx - 7) : 0;
  const int j1 = (xx < WP - 1) ? xx : (WP - 1);
  const float* Wb = Wgt + (size_t)b * HP * WP;
  float s = 0.0f;
  for (int i = i0; i <= i1; ++i) {
    const float* row = Wb + (size_t)i * WP;
    for (int j = j0; j <= j1; ++j) s += row[j];
  }
  out[idx] = E[idx] * s;
}

// ---------------------------------------------------------------------------
extern "C" void kernel_launch(void* const* d_in, const int* in_sizes, int n_in,
                              void* d_out, int out_size, void* d_ws, size_t ws_size,
                              hipStream_t stream) {
  (void)in_sizes; (void)n_in; (void)out_size; (void)ws_size;
  const float* x   = (const float*)d_in[0];   // [8,128,256,256] f32
  const float* lc  = (const float*)d_in[1];   // [8,1,249,249]   f32
  float*       out = (float*)d_out;           // [8,1,256,256]   f32

  float* E   = (float*)d_ws;                  // 8*65536  floats (2 MB)
  float* Wgt = E + (size_t)NB * HW;           // 8*249*249 floats (~2 MB)

  // Pass 1: 268 MB stream -> E = exp(channel mean)
  mean_exp_kernel<<<NB * HW / 256, 256, 0, stream>>>(x, E);

  // Pass 2: window denominators + count scaling
  const int n2 = NB * HP * WP;
  window_weight_kernel<<<(n2 + 255) / 256, 256, 0, stream>>>(E, lc, Wgt);

  // Pass 3: fold back to [8,1,256,256]
  const int n3 = NB * HW;
  fold_kernel<<<(n3 + 255) / 256, 256, 0, stream>>>(E, Wgt, out);
}